// NonstationaryAnisotropicMaternC2SpectralMixtureKernel_14534169329752
// MI455X (gfx1250) — compile-verified
//
#include <hip/hip_runtime.h>
#include <math.h>

#define ND 3
#define Q 8
#define LAT 64
#define TRI 6
#define KPAIRS 12   // 12 float2 K-pairs: 0..9 -> r-GEMM (K=20), 10 -> (w*cos, w*sin), 11 -> (0,0)

typedef float v2f __attribute__((ext_vector_type(2)));
typedef float v8f __attribute__((ext_vector_type(8)));

__device__ __forceinline__ float selu_f(float v) {
    const float scale = 1.0507009873554805f;
    const float alpha = 1.6732632423543772f;
    return v > 0.0f ? scale * v : scale * alpha * (__expf(v) - 1.0f);
}

__device__ __forceinline__ float softplus_f(float v) {
    return fmaxf(v, 0.0f) + log1pf(__expf(-fabsf(v)));
}

// ---------------------------------------------------------------------------
// Kernel 1: per-point features -> K-pair-interleaved factor arrays
//   F[q][kpair][point] is a float2 holding K-slots {2*kpair, 2*kpair+1}.
//   X slots: 0:c 1..3:u=-2Ax 4..9:a6 | 10:1 11..13:x 14..19:xO | 20:cw 21:sw 22,23:0
//   Y slots: 0:1 1..3:y      4..9:yO | 10:c 11..13:u 14..19:a6 | 20:cw 21:sw 22,23:0
//   r[i,j,q]  = sum_{k<20} SX[q,k,i]*SY[q,k,j]
//   wc[i,j,q] = sum_{k=20..23} SX[q,k,i]*SY[q,k,j]
// ---------------------------------------------------------------------------
__global__ void feature_kernel(const float* __restrict__ x, const float* __restrict__ y,
                               const float* __restrict__ W1, const float* __restrict__ b1,
                               const float* __restrict__ Ww, const float* __restrict__ bw,
                               const float* __restrict__ Wf, const float* __restrict__ bf,
                               const float* __restrict__ Ws, const float* __restrict__ bs,
                               float2* __restrict__ FX, float2* __restrict__ FY,
                               int N, int M)
{
    int t = blockIdx.x * blockDim.x + threadIdx.x;
    if (t >= N + M) return;
    bool isX = (t < N);
    int idx = isX ? t : (t - N);
    int npts = isX ? N : M;
    const float* p = isX ? (x + 3 * idx) : (y + 3 * idx);
    float2* F = isX ? FX : FY;

    float p0 = p[0], p1 = p[1], p2 = p[2];

    // hidden layer
    float h[LAT];
    #pragma unroll 8
    for (int l = 0; l < LAT; ++l) {
        float a = fmaf(W1[l * 3 + 0], p0,
                  fmaf(W1[l * 3 + 1], p1,
                  fmaf(W1[l * 3 + 2], p2, b1[l])));
        h[l] = selu_f(a);
    }

    float pO[6] = { p0 * p0, p1 * p1, p2 * p2, p0 * p1, p0 * p2, p1 * p2 };

    for (int q = 0; q < Q; ++q) {
        float aw = bw[q];
        const float* wr = Ww + q * LAT;
        #pragma unroll 8
        for (int l = 0; l < LAT; ++l) aw = fmaf(wr[l], h[l], aw);
        float w = softplus_f(aw);

        float fv[3];
        #pragma unroll
        for (int d = 0; d < 3; ++d) {
            float af = bf[q * 3 + d];
            const float* fr = Wf + (q * 3 + d) * LAT;
            #pragma unroll 8
            for (int l = 0; l < LAT; ++l) af = fmaf(fr[l], h[l], af);
            fv[d] = softplus_f(af);
        }

        float sv[6];
        #pragma unroll
        for (int ti = 0; ti < 6; ++ti) {
            float as = bs[q * 6 + ti];
            const float* sr = Ws + (q * 6 + ti) * LAT;
            #pragma unroll 8
            for (int l = 0; l < LAT; ++l) as = fmaf(sr[l], h[l], as);
            sv[ti] = softplus_f(as);
        }
        float L00 = sv[0], L10 = sv[1], L11 = sv[2], L20 = sv[3], L21 = sv[4], L22 = sv[5];
        float A00 = L00 * L00;
        float A10 = L10 * L00;
        float A11 = L10 * L10 + L11 * L11;
        float A20 = L20 * L00;
        float A21 = L20 * L10 + L21 * L11;
        float A22 = L20 * L20 + L21 * L21 + L22 * L22;

        float a6[6] = { A00, A11, A22, 2.0f * A10, 2.0f * A20, 2.0f * A21 };
        float u0 = -2.0f * (A00 * p0 + A10 * p1 + A20 * p2);
        float u1 = -2.0f * (A10 * p0 + A11 * p1 + A21 * p2);
        float u2 = -2.0f * (A20 * p0 + A21 * p1 + A22 * p2);
        float c = a6[0] * pO[0] + a6[1] * pO[1] + a6[2] * pO[2]
                + a6[3] * pO[3] + a6[4] * pO[4] + a6[5] * pO[5];

        float ph = fv[0] * p0 + fv[1] * p1 + fv[2] * p2;
        float sn, cs;
        __sincosf(6.283185307179586f * ph, &sn, &cs);

        float slots[2 * KPAIRS];
        if (isX) {
            slots[0] = c;  slots[1] = u0; slots[2] = u1; slots[3] = u2;
            #pragma unroll
            for (int k = 0; k < 6; ++k) slots[4 + k] = a6[k];
            slots[10] = 1.0f; slots[11] = p0; slots[12] = p1; slots[13] = p2;
            #pragma unroll
            for (int k = 0; k < 6; ++k) slots[14 + k] = pO[k];
        } else {
            slots[0] = 1.0f; slots[1] = p0; slots[2] = p1; slots[3] = p2;
            #pragma unroll
            for (int k = 0; k < 6; ++k) slots[4 + k] = pO[k];
            slots[10] = c; slots[11] = u0; slots[12] = u1; slots[13] = u2;
            #pragma unroll
            for (int k = 0; k < 6; ++k) slots[14 + k] = a6[k];
        }
        slots[20] = w * cs;
        slots[21] = w * sn;
        slots[22] = 0.0f;
        slots[23] = 0.0f;

        #pragma unroll
        for (int kp = 0; kp < KPAIRS; ++kp) {
            float2 v;
            v.x = slots[2 * kp + 0];
            v.y = slots[2 * kp + 1];
            F[((size_t)q * KPAIRS + kp) * npts + idx] = v;  // b64 store, lane-coalesced
        }
    }
}

// ---------------------------------------------------------------------------
// Kernel 2: pairwise via V_WMMA_F32_16X16X4_F32, 16(i) x 32(j) tile per wave.
// A 16x4 f32 fragment: lanes 0-15 hold K={kk,kk+1}, lanes 16-31 K={kk+2,kk+3}
// -> pair index = 2*step + hi, so a single per-lane float2 offset
//    (hi*npts + point) serves every load; plane strides are compile-time
//    immediates when NN/MM are template constants.
// C/D tile: element (M = v + 8*hi, N = lane&15).
// ---------------------------------------------------------------------------
template<int NN, int MM>
__global__ __launch_bounds__(256)
void pairwise_kernel(const float2* __restrict__ FX, const float2* __restrict__ FY,
                     float* __restrict__ out, int Narg, int Marg)
{
    constexpr bool S = (NN > 0);
    const int N = S ? NN : Narg;
    const int M = S ? MM : Marg;

    const int lane = threadIdx.x & 31;
    const int wave = threadIdx.x >> 5;
    const int jt = blockIdx.x;               // 32-wide j macro tile
    const int it = blockIdx.y * 8 + wave;    // 16-high i tile per wave
    const int i0 = it * 16;
    const int j0 = jt * 32;
    if (i0 >= N || j0 >= M) return;          // wave-uniform: EXEC stays all-ones

    const int col = lane & 15;
    const int hi  = lane >> 4;
    // per-lane float2-element offsets (constant across all loads)
    const int aoff  = hi * N + min(i0 + col, N - 1);
    const int boff0 = hi * M + min(j0 + col, M - 1);
    const int boff1 = hi * M + min(j0 + 16 + col, M - 1);

    const float sq5 = 2.2360679774997896f;
    const float c53 = 1.6666666666666667f;

    float acc0[8] = {0.f,0.f,0.f,0.f,0.f,0.f,0.f,0.f};
    float acc1[8] = {0.f,0.f,0.f,0.f,0.f,0.f,0.f,0.f};

    #pragma unroll
    for (int q = 0; q < Q; ++q) {
        const float2* fx = FX + (size_t)q * KPAIRS * N;
        const float2* fy = FY + (size_t)q * KPAIRS * M;

        v8f r0 = {}, r1 = {};
        #pragma unroll
        for (int s = 0; s < 5; ++s) {
            v2f a  = *reinterpret_cast<const v2f*>(fx + (size_t)(2 * s) * N + aoff);
            v2f b0 = *reinterpret_cast<const v2f*>(fy + (size_t)(2 * s) * M + boff0);
            v2f b1 = *reinterpret_cast<const v2f*>(fy + (size_t)(2 * s) * M + boff1);
            r0 = __builtin_amdgcn_wmma_f32_16x16x4_f32(false, a, false, b0,
                                                       (short)0, r0, false, false);
            r1 = __builtin_amdgcn_wmma_f32_16x16x4_f32(false, a, false, b1,
                                                       (short)0, r1, false, false);
        }

        // weight*cos tile: pairs 10 (cw,sw) + 11 (zeros)
        v2f aw  = *reinterpret_cast<const v2f*>(fx + (size_t)10 * N + aoff);
        v2f bw0 = *reinterpret_cast<const v2f*>(fy + (size_t)10 * M + boff0);
        v2f bw1 = *reinterpret_cast<const v2f*>(fy + (size_t)10 * M + boff1);
        v8f wc0 = {}, wc1 = {};
        wc0 = __builtin_amdgcn_wmma_f32_16x16x4_f32(false, aw, false, bw0,
                                                    (short)0, wc0, false, false);
        wc1 = __builtin_amdgcn_wmma_f32_16x16x4_f32(false, aw, false, bw1,
                                                    (short)0, wc1, false, false);

        // Matern-C2 envelope + q-reduction, both subtiles
        #pragma unroll
        for (int v = 0; v < 8; ++v) {
            float rr0  = r0[v];
            float pol0 = fmaf(c53 * rr0, rr0, fmaf(sq5, rr0, 1.0f));
            acc0[v] = fmaf(wc0[v], pol0 * __expf(-sq5 * rr0), acc0[v]);
            float rr1  = r1[v];
            float pol1 = fmaf(c53 * rr1, rr1, fmaf(sq5, rr1, 1.0f));
            acc1[v] = fmaf(wc1[v], pol1 * __expf(-sq5 * rr1), acc1[v]);
        }
    }

    #pragma unroll
    for (int v = 0; v < 8; ++v) {
        int row = i0 + v + 8 * hi;
        if (row < N) {
            int c0 = j0 + col;
            int c1 = j0 + 16 + col;
            if (c0 < M) out[(size_t)row * M + c0] = acc0[v];
            if (c1 < M) out[(size_t)row * M + c1] = acc1[v];
        }
    }
}

extern "C" void kernel_launch(void* const* d_in, const int* in_sizes, int n_in,
                              void* d_out, int out_size, void* d_ws, size_t ws_size,
                              hipStream_t stream) {
    const float* x  = (const float*)d_in[0];
    const float* y  = (const float*)d_in[1];
    const float* W1 = (const float*)d_in[2];
    const float* b1 = (const float*)d_in[3];
    const float* Ww = (const float*)d_in[4];
    const float* bw = (const float*)d_in[5];
    const float* Wf = (const float*)d_in[6];
    const float* bf = (const float*)d_in[7];
    const float* Ws = (const float*)d_in[8];
    const float* bs = (const float*)d_in[9];
    float* out = (float*)d_out;

    const int N = in_sizes[0] / ND;
    const int M = in_sizes[1] / ND;

    float2* FX = (float2*)d_ws;
    float2* FY = FX + (size_t)Q * KPAIRS * N;

    int total = N + M;
    feature_kernel<<<dim3((total + 255) / 256), dim3(256), 0, stream>>>(
        x, y, W1, b1, Ww, bw, Wf, bf, Ws, bs, FX, FY, N, M);

    dim3 grid((M + 31) / 32, (N + 127) / 128);
    if (N == 2048 && M == 2048) {
        pairwise_kernel<2048, 2048><<<grid, dim3(256), 0, stream>>>(FX, FY, out, N, M);
    } else {
        pairwise_kernel<0, 0><<<grid, dim3(256), 0, stream>>>(FX, FY, out, N, M);
    }
}